// GCNLayer_23751169147048
// MI455X (gfx1250) — compile-verified
//
#include <hip/hip_runtime.h>
#include <hip/hip_bf16.h>

typedef __attribute__((ext_vector_type(2))) float v2f;
typedef __attribute__((ext_vector_type(8))) float v8f;

#define D_FEAT 96

// out[r] = sum_{e: rows[e]==r} vals[e] * embeds[cols[e]]   (rows sorted)
//
// One block = 16 consecutive output rows; 6 waves/block, wave w owns feature
// columns [16w,16w+16). Edge range for the 16-row group is found by binary
// search, then consumed 16 edges/iter as 4x V_WMMA_F32_16X16X4_F32:
//   C(16x16) += A(16x4) * B(4x16)
//   A[m][k] = vals[e_k] * (rows[e_k]==row_base+m)      (selection * value)
//   B[k][n] = embeds[cols[e_k]*96 + 16w + n]           (coalesced 64B gather)
// f32 WMMA operand layout (ISA 7.12.2): lane half holds K=2*khalf (VGPR0) and
// K=2*khalf+1 (VGPR1); M/N = lane&15.
//
// Because rows is sorted, edges outside [start,end) fail the row match and
// contribute A=0 automatically — the hot loop needs NO bounds guards, only a
// clamped index near the global tail of the arrays. EXEC stays all-ones
// (WMMA requirement) and control flow is wave-uniform.
__global__ __launch_bounds__(192) void spmm_wmma_kernel(
    const int* __restrict__ rows, const int* __restrict__ cols,
    const float* __restrict__ vals, const float* __restrict__ embeds,
    float* __restrict__ out, int n_edges, int n_nodes) {
  const int g        = blockIdx.x;
  const int wave     = threadIdx.x >> 5;   // feature chunk 0..5
  const int lane     = threadIdx.x & 31;
  const int m        = lane & 15;          // M (A) / N (B,C) index
  const int khalf    = lane >> 4;          // which K pair this lane holds
  const int row_base = g * 16;
  const int rbm      = row_base + m;       // row this lane selects for A
  const int fbase    = wave * 16 + m;      // feature column this lane gathers

  // rows[] sorted: edge range [start,end) for rows [row_base, row_base+16).
  int lo = 0, hi = n_edges;
  while (lo < hi) { int mid = (lo + hi) >> 1; if (rows[mid] < row_base)      lo = mid + 1; else hi = mid; }
  const int start = lo;
  hi = n_edges;
  while (lo < hi) { int mid = (lo + hi) >> 1; if (rows[mid] < row_base + 16) lo = mid + 1; else hi = mid; }
  const int end = lo;

  const int2*   rows2 = (const int2*)rows;   // pair loads: pair start is even
  const int2*   cols2 = (const int2*)cols;
  const float2* vals2 = (const float2*)vals;
  const int pmax = (n_edges - 2) & ~1;       // last full even-aligned pair

  v8f acc = {};  // 16x16 f32 C tile

  for (int base = start & ~15; base < end; base += 16) {
    // Batch phase 1: edge metadata for 4 K-steps (b64 pair loads, broadcast
    // within each 16-lane half — cheap from L0).
    int   e0s[4];
    int2  rr[4], cc[4];
    float2 vv[4];
#pragma unroll
    for (int s = 0; s < 4; ++s) {
      const int e0 = base + 4 * s + 2 * khalf;   // even
      e0s[s] = e0;
      const int p = (e0 < pmax ? e0 : pmax) >> 1;  // clamped, stays aligned
      rr[s] = rows2[p];
      cc[s] = cols2[p];
      vv[s] = vals2[p];
    }
    // Batch phase 2: 8 embeds gathers (64B-coalesced per 16-lane half).
    float b0[4], b1[4];
#pragma unroll
    for (int s = 0; s < 4; ++s) {
      b0[s] = embeds[cc[s].x * D_FEAT + fbase];
      b1[s] = embeds[cc[s].y * D_FEAT + fbase];
    }
    // Phase 3: build A via selects (no exec divergence) + 4 WMMAs.
#pragma unroll
    for (int s = 0; s < 4; ++s) {
      const bool ok = e0s[s] <= pmax;  // both pair elements globally in-range
      const float a0 = (ok && rr[s].x == rbm) ? vv[s].x : 0.f;
      const float a1 = (ok && rr[s].y == rbm) ? vv[s].y : 0.f;
      v2f A; A.x = a0;    A.y = a1;
      v2f B; B.x = b0[s]; B.y = b1[s];
      acc = __builtin_amdgcn_wmma_f32_16x16x4_f32(false, A, false, B,
                                                  (short)0, acc, false, false);
    }
  }

  // Odd global edge count: final edge isn't covered by pair loads. One masked
  // WMMA (K=0 slot, zeros elsewhere); no-op for all but the owning group.
  if (n_edges & 1) {
    const int e = n_edges - 1;
    const int  r = rows[e];
    float a0 = 0.f, b0 = 0.f;
    if (khalf == 0) {                       // lanes holding K=0
      b0 = embeds[cols[e] * D_FEAT + fbase];
      a0 = (r == rbm) ? vals[e] : 0.f;
    }
    v2f A; A.x = a0; A.y = 0.f;
    v2f B; B.x = b0; B.y = 0.f;
    acc = __builtin_amdgcn_wmma_f32_16x16x4_f32(false, A, false, B,
                                                (short)0, acc, false, false);
  }

  // C layout: VGPR i -> M = i + 8*khalf, N = lane&15.
#pragma unroll
  for (int i = 0; i < 8; ++i) {
    const int r = row_base + i + khalf * 8;
    if (r < n_nodes)
      out[r * D_FEAT + fbase] = acc[i];
  }
}

extern "C" void kernel_launch(void* const* d_in, const int* in_sizes, int n_in,
                              void* d_out, int out_size, void* d_ws, size_t ws_size,
                              hipStream_t stream) {
  const int*   rows   = (const int*)d_in[0];
  const int*   cols   = (const int*)d_in[1];
  const float* vals   = (const float*)d_in[2];
  const float* embeds = (const float*)d_in[3];
  float*       out    = (float*)d_out;

  const int n_edges = in_sizes[0];
  const int n_nodes = out_size / D_FEAT;
  const int groups  = (n_nodes + 15) / 16;

  hipLaunchKernelGGL(spmm_wmma_kernel, dim3(groups), dim3(192), 0, stream,
                     rows, cols, vals, embeds, out, n_edges, n_nodes);
}